// SimpleSS2D_39745627357375
// MI455X (gfx1250) — compile-verified
//
#include <hip/hip_runtime.h>
#include <hip/hip_bf16.h>
#include <cstdint>

#define D_MODEL 96
#define D_STATE 16
#define D_INNER 192
#define DT_RANK 6
#define KDIR 4
#define BSZ 2
#define HH 64
#define WW 64
#define LL (HH*WW)                    // 4096
#define PP (DT_RANK + 2*D_STATE)      // 38
#define KP (KDIR*PP)                  // 152
#define KD (KDIR*D_INNER)             // 768
#define KR (KDIR*DT_RANK)             // 24

typedef float v2f __attribute__((ext_vector_type(2)));
typedef float v8f __attribute__((ext_vector_type(8)));

__device__ __forceinline__ float silu_f(float x) { return x / (1.f + __expf(-x)); }
__device__ __forceinline__ float softplus_f(float x) {
    return (x > 20.f) ? x : log1pf(__expf(x));
}

// ---------------------------------------------------------------------------
// Generic fp32 WMMA GEMM: C[M, N] = A[M, Kd] * B (+bias).
// B is (Kd x N) row-major if !BT, else stored as (N x Kd) row-major (B^T).
// BIASMODE: 0 = none, 1 = per-row bias[m], 2 = per-col bias[n].
// NT: 16-wide N tiles per wave (A fragment reused NT times per k-step,
//     NT independent WMMA accumulator chains for ILP).
// MEDGE: true if M is not a multiple of 16. Out-of-range A rows are handled
//     by ADDRESS CLAMPING (row m of A only affects row m of C, so duplicate
//     reads are harmless) -- no EXEC-mask predication in the inner loop.
// Fragment layouts (32-bit WMMA, wave32):
//   A 16x4: lane<16 -> K = k,k+1 ; lane>=16 -> K = k+2,k+3 ; M = lane&15
//   B 4x16: same K split; N = lane&15
//   C/D:    VGPR r -> M = r + 8*(lane>=16), N = lane&15
// ---------------------------------------------------------------------------
template<bool BT, int BIASMODE, int NT, bool MEDGE>
__global__ __launch_bounds__(32)
void gemm_wmma(const float* __restrict__ A, const float* __restrict__ B,
               const float* __restrict__ bias, float* __restrict__ C,
               int M, int N, int Kd, int lda, int ldb, int ldc,
               long sA, long sB, long sC)
{
    const int lane = threadIdx.x;
    const int n0 = blockIdx.x * (16 * NT);
    const int m0 = blockIdx.y * 16;
    A += (long)blockIdx.z * sA;
    B += (long)blockIdx.z * sB;
    C += (long)blockIdx.z * sC;

    const int half = (lane >= 16) ? 2 : 0;
    int mr = m0 + (lane & 15);
    if (MEDGE) mr = (mr < M) ? mr : (M - 1);   // clamp: no exec juggling
    const int nc = n0 + (lane & 15);

    v8f acc[NT];
    #pragma unroll
    for (int j = 0; j < NT; ++j) acc[j] = (v8f){};

    for (int k = 0; k < Kd; k += 4) {
        v2f a;
        {
            const float* ap = A + (size_t)mr * lda + k + half;
            a.x = ap[0]; a.y = ap[1];
        }
        #pragma unroll
        for (int j = 0; j < NT; ++j) {
            v2f b;
            if (BT) {
                const float* bp = B + (size_t)(nc + 16 * j) * ldb + k + half;
                b.x = bp[0]; b.y = bp[1];
            } else {
                const float* bp = B + (size_t)(k + half) * ldb + nc + 16 * j;
                b.x = bp[0]; b.y = bp[ldb];
            }
            acc[j] = __builtin_amdgcn_wmma_f32_16x16x4_f32(
                false, a, false, b, (short)0, acc[j], false, false);
        }
    }

    #pragma unroll
    for (int j = 0; j < NT; ++j) {
        const int ncj = nc + 16 * j;
        #pragma unroll
        for (int r = 0; r < 8; ++r) {
            int row = m0 + r + ((lane >= 16) ? 8 : 0);
            float v = acc[j][r];
            if (BIASMODE == 1) v += bias[row < M ? row : 0];
            else if (BIASMODE == 2) v += bias[ncj];
            if (!MEDGE || row < M)
                C[(size_t)row * ldc + ncj] = v;
        }
    }
}

// z = silu(xz[:, 192:384]) as (B,L,192) contiguous
__global__ void k_split_z(const float* __restrict__ xz, float* __restrict__ z)
{
    size_t i = (size_t)blockIdx.x * blockDim.x + threadIdx.x;  // B*L*192
    size_t bl = i / D_INNER;
    int d = (int)(i % D_INNER);
    z[i] = silu_f(xz[bl * (2 * D_INNER) + D_INNER + d]);
}

// xx[b,d,l] = xz[b,l,d] (first 192 channels), via LDS tile transpose
__global__ void k_xx_transpose(const float* __restrict__ xz, float* __restrict__ xx)
{
    __shared__ float tile[32][33];
    int b = blockIdx.z;
    int l0 = blockIdx.x * 32;
    int d0 = blockIdx.y * 32;
    int tx = threadIdx.x, ty = threadIdx.y;  // 32 x 8
    #pragma unroll
    for (int r = 0; r < 4; ++r) {
        int l = l0 + ty + r * 8;
        tile[ty + r * 8][tx] = xz[((size_t)b * LL + l) * (2 * D_INNER) + d0 + tx];
    }
    __syncthreads();
    #pragma unroll
    for (int r = 0; r < 4; ++r) {
        int d = d0 + ty + r * 8;
        xx[((size_t)b * D_INNER + d) * LL + l0 + tx] = tile[tx][ty + r * 8];
    }
}

// depthwise 3x3 SAME conv + bias + silu on (B,192,64,64)
__global__ void k_dwconv_silu(const float* __restrict__ xx, const float* __restrict__ cw,
                              const float* __restrict__ cb, float* __restrict__ xc)
{
    int i = blockIdx.x * blockDim.x + threadIdx.x;  // B*D*H*W
    int w = i & (WW - 1);
    int t = i >> 6;
    int h = t & (HH - 1);
    t >>= 6;
    int d = t % D_INNER;
    int b = t / D_INNER;
    const float* base = xx + ((size_t)b * D_INNER + d) * LL;
    const float* wk = cw + d * 9;
    float s = 0.f;
    #pragma unroll
    for (int di = -1; di <= 1; ++di) {
        #pragma unroll
        for (int dj = -1; dj <= 1; ++dj) {
            int h2 = h + di, w2 = w + dj;
            if (h2 >= 0 && h2 < HH && w2 >= 0 && w2 < WW)
                s += base[h2 * WW + w2] * wk[(di + 1) * 3 + (dj + 1)];
        }
    }
    xc[i] = silu_f(s + cb[d]);
}

// xs[b,k,d,l]: k0 row-major, k1 col-major, k2/k3 reversed
__global__ void k_cross_scan(const float* __restrict__ xc, float* __restrict__ xs)
{
    size_t i = (size_t)blockIdx.x * blockDim.x + threadIdx.x;  // B*K*D*L
    int l = (int)(i & (LL - 1));
    size_t t = i >> 12;
    int d = (int)(t % D_INNER);
    t /= D_INNER;
    int k = (int)(t & 3);
    int b = (int)(t >> 2);
    int pos;
    if (k == 0) pos = l;
    else if (k == 2) pos = LL - 1 - l;
    else {
        int lr = (k == 1) ? l : (LL - 1 - l);
        pos = (lr & (HH - 1)) * WW + (lr >> 6);   // (l%H)*W + l/H
    }
    xs[i] = xc[((size_t)b * D_INNER + d) * LL + pos];
}

// dts[b,r,l] = x_dbl[b, (r/6)*38 + r%6, l]
__global__ void k_extract_dts(const float* __restrict__ xdbl, float* __restrict__ dts)
{
    size_t i = (size_t)blockIdx.x * blockDim.x + threadIdx.x;  // B*24*L
    int l = (int)(i & (LL - 1));
    size_t t = i >> 12;
    int r = (int)(t % KR);
    int b = (int)(t / KR);
    int k = r / DT_RANK, rr = r % DT_RANK;
    dts[i] = xdbl[((size_t)b * KP + k * PP + rr) * LL + l];
}

// Chunked associative selective scan. One wave per (b,k,d) sequence; each of
// 32 lanes owns a 128-step chunk. Chunk transition multiplier per state n is
// exp(A_n * sum(dt)) -- analytic product of per-step exp(dt*A_n). Affine
// (a,b) Hillis-Steele scan over lanes gives chunk-entry states. Phase 3
// replays chunks and writes y (in-place over the dt buffer).
__global__ __launch_bounds__(256)
void k_selective_scan(const float* __restrict__ xs, const float* __restrict__ xdbl,
                      const float* __restrict__ A_log, const float* __restrict__ Dsk,
                      const float* __restrict__ dt_bias, float* __restrict__ dtbuf)
{
    int wid = (blockIdx.x * blockDim.x + threadIdx.x) >> 5;  // 0..B*K*D-1
    int lane = threadIdx.x & 31;
    int d = wid % D_INNER;
    int t = wid / D_INNER;
    int k = t & 3;
    int b = t >> 2;
    int kd = k * D_INNER + d;

    const float* up = xs + (size_t)wid * LL;
    float* dtp = dtbuf + (size_t)wid * LL;
    const float* Bp = xdbl + ((size_t)b * KP + k * PP + DT_RANK) * LL;
    const float* Cp = Bp + (size_t)D_STATE * LL;

    float Avec[16];
    #pragma unroll
    for (int n = 0; n < 16; ++n) Avec[n] = -__expf(A_log[(size_t)kd * 16 + n]);
    const float dtb = dt_bias[kd];
    const float dskip = Dsk[kd];

    const int CH = LL / 32;  // 128
    const int l0 = lane * CH;

    // phase 1: per-chunk aggregate (sum dt, local h accumulator from h=0)
    float sdt = 0.f;
    float hb[16];
    #pragma unroll
    for (int n = 0; n < 16; ++n) hb[n] = 0.f;
    for (int i = 0; i < CH; ++i) {
        int l = l0 + i;
        float dtv = softplus_f(dtp[l] + dtb);
        float xdt = dtv * up[l];
        sdt += dtv;
        #pragma unroll
        for (int n = 0; n < 16; ++n) {
            float dA = __expf(dtv * Avec[n]);
            hb[n] = dA * hb[n] + xdt * Bp[(size_t)n * LL + l];
        }
    }
    float aA[16];
    #pragma unroll
    for (int n = 0; n < 16; ++n) aA[n] = __expf(sdt * Avec[n]);

    // phase 2: inclusive affine scan over chunks:
    // (a,b)_i = (a_i * a_prev, a_i * b_prev + b_i)
    #pragma unroll
    for (int off = 1; off < 32; off <<= 1) {
        #pragma unroll
        for (int n = 0; n < 16; ++n) {
            float pa = __shfl_up(aA[n], (unsigned)off, 32);
            float pb = __shfl_up(hb[n], (unsigned)off, 32);
            if (lane >= off) {
                hb[n] = aA[n] * pb + hb[n];
                aA[n] *= pa;
            }
        }
    }
    float h[16];
    #pragma unroll
    for (int n = 0; n < 16; ++n) {
        float ph = __shfl_up(hb[n], 1u, 32);
        h[n] = (lane == 0) ? 0.f : ph;
    }

    // phase 3: replay chunk from entry state, emit y = <h,C> + D*u (in place)
    for (int i = 0; i < CH; ++i) {
        int l = l0 + i;
        float dtv = softplus_f(dtp[l] + dtb);
        float u = up[l];
        float xdt = dtv * u;
        float y = 0.f;
        #pragma unroll
        for (int n = 0; n < 16; ++n) {
            float dA = __expf(dtv * Avec[n]);
            h[n] = dA * h[n] + xdt * Bp[(size_t)n * LL + l];
            y += h[n] * Cp[(size_t)n * LL + l];
        }
        dtp[l] = y + dskip * u;
    }
}

// ym[b,l,d] = ys0[l] + ys2[L-1-l] + ys1[lt] + ys3[L-1-lt], lt = (l%64)*64 + l/64
__global__ void k_cross_merge(const float* __restrict__ ys, float* __restrict__ ym)
{
    __shared__ float tile[32][33];
    int b = blockIdx.z;
    int l0 = blockIdx.x * 32;
    int d0 = blockIdx.y * 32;
    int tx = threadIdx.x, ty = threadIdx.y;
    int l = l0 + tx;
    int lr = LL - 1 - l;
    int lt = (l & (HH - 1)) * WW + (l >> 6);
    int ltr = LL - 1 - lt;
    #pragma unroll
    for (int r = 0; r < 4; ++r) {
        int dloc = ty + r * 8;
        const float* base = ys + ((size_t)b * KDIR * D_INNER + (d0 + dloc)) * LL;
        float v = base[l]
                + base[(size_t)2 * D_INNER * LL + lr]
                + base[(size_t)1 * D_INNER * LL + lt]
                + base[(size_t)3 * D_INNER * LL + ltr];
        tile[dloc][tx] = v;
    }
    __syncthreads();
    #pragma unroll
    for (int r = 0; r < 4; ++r) {
        int lloc = ty + r * 8;
        ym[((size_t)b * LL + l0 + lloc) * D_INNER + d0 + tx] = tile[tx][lloc];
    }
}

// LayerNorm over D_INNER then gate with silu(z); in place. One wave per (b,l).
__global__ __launch_bounds__(256)
void k_ln_gate(float* __restrict__ ym, const float* __restrict__ z,
               const float* __restrict__ gamma, const float* __restrict__ beta)
{
    int wid = (blockIdx.x * blockDim.x + threadIdx.x) >> 5;  // 0..B*L-1
    int lane = threadIdx.x & 31;
    float* row = ym + (size_t)wid * D_INNER;
    const float* zrow = z + (size_t)wid * D_INNER;
    float v[6];
    float s = 0.f, s2 = 0.f;
    #pragma unroll
    for (int i = 0; i < 6; ++i) {
        v[i] = row[lane + i * 32];
        s += v[i];
        s2 += v[i] * v[i];
    }
    #pragma unroll
    for (int off = 16; off > 0; off >>= 1) {
        s += __shfl_xor(s, off, 32);
        s2 += __shfl_xor(s2, off, 32);
    }
    const float inv_n = 1.f / (float)D_INNER;
    float mu = s * inv_n;
    float var = s2 * inv_n - mu * mu;
    float inv = rsqrtf(var + 1e-5f);
    #pragma unroll
    for (int i = 0; i < 6; ++i) {
        int d = lane + i * 32;
        float val = (v[i] - mu) * inv * gamma[d] + beta[d];
        row[d] = val * zrow[d];
    }
}

extern "C" void kernel_launch(void* const* d_in, const int* in_sizes, int n_in,
                              void* d_out, int out_size, void* d_ws, size_t ws_size,
                              hipStream_t stream)
{
    const float* x       = (const float*)d_in[0];
    const float* W_in    = (const float*)d_in[1];
    const float* b_in    = (const float*)d_in[2];
    const float* conv_w  = (const float*)d_in[3];
    const float* conv_b  = (const float*)d_in[4];
    const float* W_xp    = (const float*)d_in[5];
    const float* b_xp    = (const float*)d_in[6];
    const float* W_dt    = (const float*)d_in[7];
    const float* b_dt    = (const float*)d_in[8];
    const float* A_log   = (const float*)d_in[9];
    const float* D_skip  = (const float*)d_in[10];
    const float* dt_bias = (const float*)d_in[11];
    const float* W_out   = (const float*)d_in[12];
    const float* b_out   = (const float*)d_in[13];
    const float* gamma   = (const float*)d_in[14];
    const float* beta    = (const float*)d_in[15];
    float* out = (float*)d_out;
    float* ws = (float*)d_ws;

    const size_t BL = (size_t)BSZ * LL;            // 8192
    float* xz   = ws;                              // BL*384     = 3,145,728
    float* xx   = xz  + BL * 2 * D_INNER;          // B*192*L    = 1,572,864
    float* z    = xx  + (size_t)BSZ * D_INNER * LL;
    float* xc   = z   + (size_t)BSZ * D_INNER * LL;
    float* xs   = xc  + (size_t)BSZ * D_INNER * LL;         // B*768*L = 6,291,456
    float* xdbl = xs  + (size_t)BSZ * KD * LL;              // B*152*L = 1,245,184
    float* dts  = xdbl + (size_t)BSZ * KP * LL;             // B*24*L  = 196,608
    float* dtb  = dts + (size_t)BSZ * KR * LL;              // B*768*L = 6,291,456
    float* ym   = dtb + (size_t)BSZ * KD * LL;              // B*L*192 = 1,572,864
    // total: ~93.8 MB of workspace

    // 1) xz = x @ W_in^T + b_in   (M=8192, N=384, K=96), NT=4 -> grid.x = 6
    gemm_wmma<true, 2, 4, false><<<dim3(2 * D_INNER / 64, BL / 16, 1), 32, 0, stream>>>(
        x, W_in, b_in, xz, (int)BL, 2 * D_INNER, D_MODEL,
        D_MODEL, D_MODEL, 2 * D_INNER, 0, 0, 0);

    // 2) split: z = silu(xz[:,192:]) ; xx = transpose(xz[:,:192]) -> (B,192,L)
    k_split_z<<<(int)(BL * D_INNER / 256), 256, 0, stream>>>(xz, z);
    k_xx_transpose<<<dim3(LL / 32, D_INNER / 32, BSZ), dim3(32, 8), 0, stream>>>(xz, xx);

    // 3) depthwise conv + silu
    k_dwconv_silu<<<(int)(BL * D_INNER / 256), 256, 0, stream>>>(xx, conv_w, conv_b, xc);

    // 4) cross scan -> xs (B,K,D,L)
    k_cross_scan<<<(int)((size_t)BSZ * KDIR * D_INNER * LL / 256), 256, 0, stream>>>(xc, xs);

    // 5) x_dbl = W_xp @ xs + b_xp, per batch  (M=152, N=4096, K=768), M-edge
    gemm_wmma<false, 1, 4, true><<<dim3(LL / 64, (KP + 15) / 16, BSZ), 32, 0, stream>>>(
        W_xp, xs, b_xp, xdbl, KP, LL, KD,
        KD, LL, LL, 0, (long)KD * LL, (long)KP * LL);

    // 6) extract dts (B,24,L)
    k_extract_dts<<<(int)((size_t)BSZ * KR * LL / 256), 256, 0, stream>>>(xdbl, dts);

    // 7) dt = W_dt @ dts + b_dt, per batch  (M=768, N=4096, K=24)
    gemm_wmma<false, 1, 4, false><<<dim3(LL / 64, KD / 16, BSZ), 32, 0, stream>>>(
        W_dt, dts, b_dt, dtb, KD, LL, KR,
        KR, LL, LL, 0, (long)KR * LL, (long)KD * LL);

    // 8) chunked selective scan (ys written in place over dtb)
    k_selective_scan<<<(BSZ * KDIR * D_INNER) / 8, 256, 0, stream>>>(
        xs, xdbl, A_log, D_skip, dt_bias, dtb);

    // 9) cross merge -> ym (B,L,192)
    k_cross_merge<<<dim3(LL / 32, D_INNER / 32, BSZ), dim3(32, 8), 0, stream>>>(dtb, ym);

    // 10) LayerNorm + silu(z) gate, in place
    k_ln_gate<<<(int)(BL / 8), 256, 0, stream>>>(ym, z, gamma, beta);

    // 11) out = ym @ W_out^T + b_out  (M=8192, N=96, K=192), NT=2 -> grid.x = 3
    gemm_wmma<true, 2, 2, false><<<dim3(D_MODEL / 32, BL / 16, 1), 32, 0, stream>>>(
        ym, W_out, b_out, out, (int)BL, D_MODEL, D_INNER,
        D_INNER, D_INNER, D_MODEL, 0, 0, 0);
}